// MHESA_14929306321231
// MI455X (gfx1250) — compile-verified
//
#include <hip/hip_runtime.h>

// MHESA: LayerNorm + per-head causal EMA scan + residual, for MI455X (gfx1250).
// 3-pass chunked-scan decomposition; intra-chunk 16x16 triangular scan runs on
// V_WMMA_F32_16X16X4_F32 (4 chained K=4 WMMAs per 16-step chunk).

#define DIMC   1024
#define HEADS_ 8
#define LSEQ   8192
#define BATCH  4
#define SEG    512
#define NSEG   (LSEQ / SEG)   // 16
#define STRIPS (DIMC / 16)    // 64 channel strips of 16
#define CHUNKS (SEG / 16)     // 32
#define LN_EPS 1e-5f

typedef __attribute__((ext_vector_type(2))) float v2f;
typedef __attribute__((ext_vector_type(8))) float v8f;

__device__ __forceinline__ float powi_f(float b, int e) {
  float r = 1.0f;
  for (int i = 0; i < e; ++i) r *= b;
  return r;
}

// ---------------- Pass 1: LayerNorm row statistics -------------------------
// One wave (32 lanes) per (b,t) row of 1024 channels. 8 waves / block.
__global__ void ln_stats_kernel(const float* __restrict__ x,
                                float2* __restrict__ stats) {
  const int wave = threadIdx.x >> 5;
  const int lane = threadIdx.x & 31;
  const int row  = blockIdx.x * 8 + wave;              // 0 .. BATCH*LSEQ-1
  const float4* xr = (const float4*)(x + (size_t)row * DIMC);
  float s1 = 0.0f, s2 = 0.0f;
#pragma unroll
  for (int i = 0; i < 8; ++i) {                        // 8 * 32 lanes * 4 = 1024
    float4 v = xr[lane + 32 * i];
    s1 += v.x + v.y + v.z + v.w;
    s2 += v.x * v.x + v.y * v.y + v.z * v.z + v.w * v.w;
  }
#pragma unroll
  for (int off = 16; off > 0; off >>= 1) {             // wave32 reduction
    s1 += __shfl_xor(s1, off, 32);
    s2 += __shfl_xor(s2, off, 32);
  }
  if (lane == 0) {
    float mean = s1 * (1.0f / DIMC);
    float var  = s2 * (1.0f / DIMC) - mean * mean;
    stats[row] = make_float2(mean, rsqrtf(var + LN_EPS));
  }
}

// ---------------- Pass 2: per-segment exit carries -------------------------
// Block = 1 wave, grid = BATCH * STRIPS * NSEG. Lane (n, hi): channel n of the
// strip, t parity hi. carry[n] = sum_t a*(1-a)^(segEnd - t) * u[t,n].
__global__ void ema_carry_kernel(const float* __restrict__ x,
                                 const float* __restrict__ gamma,
                                 const float* __restrict__ beta,
                                 const float* __restrict__ alphas,
                                 const float2* __restrict__ stats,
                                 float* __restrict__ carry) {
  const int blk   = blockIdx.x;
  const int b     = blk >> 10;           // STRIPS*NSEG = 1024 blocks per batch
  const int rem   = blk & 1023;
  const int strip = rem >> 4;
  const int seg   = rem & 15;
  const int lane  = threadIdx.x & 31;
  const int n     = lane & 15;
  const int hi    = lane >> 4;
  const int ch    = strip * 16 + n;

  const float a   = 1.0f / (1.0f + expf(-alphas[strip >> 3]));
  const float omA = 1.0f - a;
  const float d2  = omA * omA;
  const float gam = gamma[ch], bet = beta[ch];

  const int     t0 = seg * SEG;
  const float*  xb = x + (size_t)b * LSEQ * DIMC;
  const float2* st = stats + (size_t)b * LSEQ;

  float acc = 0.0f;                       // weight a*d2^(255-k) on u[t0+2k+hi]
  for (int k = 0; k < SEG / 2; ++k) {
    int    t  = t0 + 2 * k + hi;
    float  xv = xb[(size_t)t * DIMC + ch];
    float2 s  = st[t];
    float  u  = (xv - s.x) * s.y * gam + bet;
    acc = acc * d2 + a * u;
  }
  // combine parities: carry[n] = acc_odd[n] + (1-a)*acc_even[n]
  float up = __shfl(acc, n + 16, 32);
  if (lane < 16) {
    carry[(((size_t)b * STRIPS + strip) * NSEG + seg) * 16 + n] =
        up + omA * acc;
  }
}

// ---------------- Pass 3: WMMA chunked scan + residual + store -------------
__global__ void ema_scan_kernel(const float* __restrict__ x,
                                const float* __restrict__ gamma,
                                const float* __restrict__ beta,
                                const float* __restrict__ alphas,
                                const float* __restrict__ param_D,
                                const float2* __restrict__ stats,
                                const float* __restrict__ carry,
                                float* __restrict__ out) {
  const int blk   = blockIdx.x;
  const int b     = blk >> 10;
  const int rem   = blk & 1023;
  const int strip = rem >> 4;
  const int seg   = rem & 15;
  const int lane  = threadIdx.x & 31;
  const int n     = lane & 15;
  const int hi    = lane >> 4;
  const int m     = n;                    // A-matrix row index for this lane
  const int ch    = strip * 16 + n;

  const float a   = 1.0f / (1.0f + expf(-alphas[strip >> 3]));
  const float omA = 1.0f - a;
  const float gam = gamma[ch], bet = beta[ch], pD = param_D[ch];

  // A operand: 4 K-slices of lower-triangular T[m,s] = a*(1-a)^(m-s), s<=m.
  // ISA layout (32-bit A 16x4): lane m holds K = {0,1}+2*hi in VGPR {0,1}.
  v2f A[4];
#pragma unroll
  for (int q = 0; q < 4; ++q) {
    int sA = 4 * q + 2 * hi;
    int sB = sA + 1;
    A[q].x = (sA <= m) ? a * powi_f(omA, m - sA) : 0.0f;
    A[q].y = (sB <= m) ? a * powi_f(omA, m - sB) : 0.0f;
  }
  // dcol[j] = (1-a)^(mD+1) with mD = j + 8*hi (D layout row of acc[j])
  float dcol[8];
  dcol[0] = powi_f(omA, 1 + 8 * hi);
#pragma unroll
  for (int j = 1; j < 8; ++j) dcol[j] = dcol[j - 1] * omA;

  // incoming cross-segment carry: g = sum_{j<seg} ((1-a)^512)^(seg-1-j)*c_j
  float dseg = omA;
#pragma unroll
  for (int i = 0; i < 9; ++i) dseg *= dseg;            // (1-a)^512
  const float* cb = carry + ((size_t)b * STRIPS + strip) * NSEG * 16;
  float g = 0.0f;
  for (int j = 0; j < seg; ++j) g = dseg * g + cb[j * 16 + n];

  const float*  xb = x + (size_t)b * LSEQ * DIMC;
  const float2* st = stats + (size_t)b * LSEQ;
  float*        ob = out + (size_t)b * LSEQ * DIMC;

  for (int chunk = 0; chunk < CHUNKS; ++chunk) {
    const int t0 = seg * SEG + chunk * 16;

    // Load+normalize the 16x16 u tile in B layout:
    // uB[r] = u[t0 + 4*(r>>1) + (r&1) + 2*hi][ch]  (B 4x16: K=v+2*hi per reg)
    float uB[8];
#pragma unroll
    for (int r = 0; r < 8; ++r) {
      int    s  = 4 * (r >> 1) + (r & 1) + 2 * hi;
      int    t  = t0 + s;
      float  xv = xb[(size_t)t * DIMC + ch];
      float2 sv = st[t];
      uB[r] = (xv - sv.x) * sv.y * gam + bet;
    }

    // Prefetch next chunk's rows of this strip (global_prefetch_b8).
    if (chunk + 1 < CHUNKS) {
      __builtin_prefetch(&xb[(size_t)(t0 + 16 + n) * DIMC + strip * 16], 0, 1);
    }

    // C init: carry outer product (1-a)^(mD+1) * g[n]
    v8f acc;
#pragma unroll
    for (int j = 0; j < 8; ++j) acc[j] = dcol[j] * g;

    // Y = T*U + C via 4 chained f32 WMMAs (K=16 total)
    v2f b0 = {uB[0], uB[1]};
    acc = __builtin_amdgcn_wmma_f32_16x16x4_f32(false, A[0], false, b0,
                                                (short)0, acc, false, false);
    v2f b1 = {uB[2], uB[3]};
    acc = __builtin_amdgcn_wmma_f32_16x16x4_f32(false, A[1], false, b1,
                                                (short)0, acc, false, false);
    v2f b2 = {uB[4], uB[5]};
    acc = __builtin_amdgcn_wmma_f32_16x16x4_f32(false, A[2], false, b2,
                                                (short)0, acc, false, false);
    v2f b3 = {uB[6], uB[7]};
    acc = __builtin_amdgcn_wmma_f32_16x16x4_f32(false, A[3], false, b3,
                                                (short)0, acc, false, false);

    // Next-chunk carry = Y[15][n]  (D VGPR 7, lanes 16..31)
    float gNext = __shfl(acc[7], n + 16, 32);

    // Residual + store. acc[j] = Y[j + 8*hi][n]; fetch u[mD][n] from B-layout
    // regs: reg = 2*(mD>>2) + (mD&1), src lane = n + 16*((mD>>1)&1).
#pragma unroll
    for (int j = 0; j < 8; ++j) {
      const int rLo = 2 * (j >> 2) + (j & 1);          // mD = j     (hi == 0)
      const int src = n + 16 * ((j >> 1) & 1);         // same for mD = j + 8
      float uLo = __shfl(uB[rLo], src, 32);
      float uHi = __shfl(uB[rLo + 4], src, 32);        // mD = j + 8 (hi == 1)
      float um  = hi ? uHi : uLo;
      int   t   = t0 + j + 8 * hi;
      ob[(size_t)t * DIMC + ch] = acc[j] + um * pD;
    }
    g = gNext;
  }
}

// ---------------------------------------------------------------------------
extern "C" void kernel_launch(void* const* d_in, const int* in_sizes, int n_in,
                              void* d_out, int out_size, void* d_ws,
                              size_t ws_size, hipStream_t stream) {
  (void)in_sizes; (void)n_in; (void)out_size; (void)ws_size;
  const float* x       = (const float*)d_in[0];
  const float* gamma   = (const float*)d_in[1];
  const float* beta    = (const float*)d_in[2];
  const float* alphas  = (const float*)d_in[3];
  const float* param_D = (const float*)d_in[4];
  float*       out     = (float*)d_out;

  // workspace: stats float2[BATCH*LSEQ] then carry float[BATCH*STRIPS*NSEG*16]
  float2* stats = (float2*)d_ws;
  float*  carry = (float*)((char*)d_ws + sizeof(float2) * BATCH * LSEQ);

  // Pass 1: row stats (8 rows / 256-thread block)
  ln_stats_kernel<<<dim3(BATCH * LSEQ / 8), dim3(256), 0, stream>>>(x, stats);
  // Pass 2: segment carries (one wave per (b, strip, seg))
  ema_carry_kernel<<<dim3(BATCH * STRIPS * NSEG), dim3(32), 0, stream>>>(
      x, gamma, beta, alphas, stats, carry);
  // Pass 3: WMMA chunked scan + residual
  ema_scan_kernel<<<dim3(BATCH * STRIPS * NSEG), dim3(32), 0, stream>>>(
      x, gamma, beta, alphas, param_D, stats, carry, out);
}